// TreeLayer_79336635892008
// MI455X (gfx1250) — compile-verified
//
#include <hip/hip_runtime.h>

typedef float v2f __attribute__((ext_vector_type(2)));
typedef float v8f __attribute__((ext_vector_type(8)));

#define DEPTH   10
#define HEADS   16
#define DIM     256
#define NNODES  1023   // 2^10 - 1
#define NLEAF   1024

// One wave (32 lanes) per query b. Lane L: item/head h = L&15, dim-half hi = L>>4.
// Per level: 64 x v_wmma_f32_16x16x4_f32 (B = ones) reduce 16 heads x 256 dims
// of w = exp(-s_vec) into 16 totals on the matrix pipe while VALU does the exps.
__global__ __launch_bounds__(256) void tree_layer_kernel(
    const float* __restrict__ q,    // [B, 256]
    const float* __restrict__ kp,   // [16, 1023, 2, 256]
    const float* __restrict__ vp,   // [16, 1024, 256]
    float* __restrict__ out,        // [B] support  ++  [B*16, 256] val
    int Bq)
{
    const int lane = threadIdx.x & 31;
    const int wave = threadIdx.x >> 5;
    const int b    = blockIdx.x * 8 + wave;
    if (b >= Bq) return;                       // wave-uniform: EXEC stays full

    const int h  = lane & 15;
    const int hi = lane >> 4;

    const float* qrow  = q  + (size_t)b * DIM;
    const float* khead = kp + (size_t)h * (NNODES * 2 * DIM);

    v2f bones; bones[0] = 1.0f; bones[1] = 1.0f;   // B = ones(4x16)

    // lane -> D-matrix extraction constants
    const bool halfmatch = ((lane >> 4) == ((lane >> 3) & 1));
    const int  l1 = lane & 1, l2 = lane & 2, l4 = lane & 4;

    int   ix   = 0;
    float supp = 0.0f;
    const int offs = hi << 1;                  // 0 or 2: K-offset within each 4-dim tile

    for (int d = 0; d < DEPTH; ++d) {
        const int node = (1 << d) - 1 + ix;
        const float* k0 = khead + (size_t)node * (2 * DIM);
        const float* k1 = k0 + DIM;

        v8f c = {};                            // accumulator: totals for 16 heads
        #pragma unroll 4
        for (int t = 0; t < 64; ++t) {
            const int o = t * 4 + offs;        // this lane's two dims of the tile
            v2f qv  = *(const v2f*)(qrow + o);
            v2f k0v = *(const v2f*)(k0 + o);
            v2f k1v = *(const v2f*)(k1 + o);
            v2f a;
            {   // w = 1 / ( 1/(e^{-k0}+e^{-q}) + 1/(e^{-k1}+e^{q}) )  == exp(-s_vec)
                float eqm = __expf(-qv[0]);
                float eqp = __builtin_amdgcn_rcpf(eqm);
                float t1  = __builtin_amdgcn_rcpf(__expf(-k0v[0]) + eqm);
                float t2  = __builtin_amdgcn_rcpf(__expf(-k1v[0]) + eqp);
                a[0]      = __builtin_amdgcn_rcpf(t1 + t2);
            }
            {
                float eqm = __expf(-qv[1]);
                float eqp = __builtin_amdgcn_rcpf(eqm);
                float t1  = __builtin_amdgcn_rcpf(__expf(-k0v[1]) + eqm);
                float t2  = __builtin_amdgcn_rcpf(__expf(-k1v[1]) + eqp);
                a[1]      = __builtin_amdgcn_rcpf(t1 + t2);
            }
            // C[r, :] += sum_k A[r, k] : 16 head-totals accumulate on the XDL pipe
            c = __builtin_amdgcn_wmma_f32_16x16x4_f32(
                    false, a, false, bones, (short)0, c, false, false);
        }

        // D columns are identical; pick row m = lane&7, then fix the lane-half.
        float s01 = l1 ? c[1] : c[0];
        float s23 = l1 ? c[3] : c[2];
        float s45 = l1 ? c[5] : c[4];
        float s67 = l1 ? c[7] : c[6];
        float s03 = l2 ? s23 : s01;
        float s47 = l2 ? s67 : s45;
        float sel = l4 ? s47 : s03;
        float sw  = __shfl_xor(sel, 16, 32);
        float total = halfmatch ? sel : sw;    // total = sum_j exp(-s_vec_j)

        // s = -log(total); bit = (s >= 0) <=> total <= 1; support += exp(-|s|)
        float r = __builtin_amdgcn_rcpf(total);
        supp += fminf(total, r);
        int bit = (total <= 1.0f) ? 1 : 0;
        ix = 2 * ix + bit;
    }

    // value gather: lane pair (h, hi) copies its 256-float row (128 each half)
    {
        const float* vrow = vp + ((size_t)h * NLEAF + ix) * DIM + hi * 128;
        float*       orow = out + Bq + ((size_t)b * HEADS + h) * DIM + hi * 128;
        #pragma unroll
        for (int t = 0; t < 32; ++t) {
            float4 v = *(const float4*)(vrow + t * 4);
            *(float4*)(orow + t * 4) = v;
        }
    }

    // support over heads: exp(-support_h) == supp_h, so out = -log(sum_h supp_h).
    // Each head appears in two lanes -> halve the full-wave sum.
    float sum = supp;
    sum += __shfl_xor(sum, 16, 32);
    sum += __shfl_xor(sum, 8, 32);
    sum += __shfl_xor(sum, 4, 32);
    sum += __shfl_xor(sum, 2, 32);
    sum += __shfl_xor(sum, 1, 32);
    if (lane == 0) out[b] = -__logf(0.5f * sum);
}

extern "C" void kernel_launch(void* const* d_in, const int* in_sizes, int n_in,
                              void* d_out, int out_size, void* d_ws, size_t ws_size,
                              hipStream_t stream) {
    const float* q  = (const float*)d_in[0];   // [8192, 256]
    const float* kp = (const float*)d_in[1];   // [16, 1023, 2, 256]
    const float* vp = (const float*)d_in[2];   // [16, 1024, 256]
    float* out = (float*)d_out;                // [8192] ++ [131072, 256]

    const int Bq = in_sizes[0] / DIM;          // 8192
    const int waves_per_block = 8;             // 256 threads = 8 wave32
    const int nblocks = (Bq + waves_per_block - 1) / waves_per_block;

    tree_layer_kernel<<<nblocks, 256, 0, stream>>>(q, kp, vp, out, Bq);
}